// CTC_45621142618733
// MI455X (gfx1250) — compile-verified
//
#include <hip/hip_runtime.h>
#include <hip/hip_bf16.h>
#include <cstdint>

typedef __attribute__((ext_vector_type(2))) float v2f;
typedef __attribute__((ext_vector_type(8))) float v8f;

typedef __attribute__((address_space(1))) int glb_i32;
typedef __attribute__((address_space(3))) int lds_i32;

#define B_  32
#define T_  2000
#define C_  1024
#define L_  100
#define S_  201        // 2L+1 lattice states
#define NEG (-1e30f)
#define NB  8          // async prefetch depth (power of two)

// ---------------------------------------------------------------------------
// async copy helpers (CDNA5 async-to-LDS path, guarded)
// ---------------------------------------------------------------------------
__device__ __forceinline__ void async_copy_b32(const float* gsrc, float* ldst) {
#if __has_builtin(__builtin_amdgcn_global_load_async_to_lds_b32)
  __builtin_amdgcn_global_load_async_to_lds_b32(
      (glb_i32*)(gsrc),   // addrspace(1) int*, const cast away
      (lds_i32*)(ldst),   // addrspace(3) int*
      /*offset=*/0, /*cpol=*/0);
#else
  *ldst = *gsrc;  // synchronous fallback: plain load + ds_store
#endif
}

template <int N>
__device__ __forceinline__ void wait_asynccnt_le() {
#if __has_builtin(__builtin_amdgcn_s_wait_asynccnt)
  __builtin_amdgcn_s_wait_asynccnt((short)N);
#else
  asm volatile("s_wait_asynccnt %0" ::"i"(N));
#endif
}

// ---------------------------------------------------------------------------
// Kernel 1: lse[r] = logsumexp over C of inputs row r, r in [0, B*T).
// One wave handles 16 rows; exp-sums computed with V_WMMA_F32_16X16X4_F32
// against a ones B-matrix (D[m][n] = sum_k A[m][k] for every n).
// ---------------------------------------------------------------------------
__global__ __launch_bounds__(256) void lse_wmma_kernel(
    const float* __restrict__ x, float* __restrict__ lse) {
  const int tid  = threadIdx.x;
  const int lane = tid & 31;
  const int wave = tid >> 5;
  const int R    = (blockIdx.x * 8 + wave) * 16;  // first of 16 rows
  const int m    = lane & 15;                     // row within tile
  const int h    = lane >> 4;                     // half-wave id

  const float* rp = x + (size_t)(R + m) * C_;

  // ---- phase 1: row max (each half-wave covers half the columns) ----
  const float4* rp4 = reinterpret_cast<const float4*>(rp + h * (C_ / 2));
  float mx = -__builtin_inff();
#pragma unroll 4
  for (int j = 0; j < C_ / 8; ++j) {
    float4 v = rp4[j];
    mx = fmaxf(mx, fmaxf(fmaxf(v.x, v.y), fmaxf(v.z, v.w)));
  }
  mx = fmaxf(mx, __shfl_xor(mx, 16, 32));  // lane m and m+16 now hold row max

  // ---- phase 2: sum of exp via WMMA f32 16x16x4, B = ones ----
  // A layout (16x4 f32): VGPR0 = {h=0: K=0, h=1: K=2}, VGPR1 = {K=1, K=3}
  v2f bones;  bones[0] = 1.0f;  bones[1] = 1.0f;
  v8f acc0 = {};
  v8f acc1 = {};
  const v2f* rp2 = reinterpret_cast<const v2f*>(rp);
#pragma unroll 4
  for (int k = 0; k < C_; k += 8) {
    v2f v0 = rp2[(k >> 1) + h];          // cols k+2h, k+2h+1
    v2f v1 = rp2[((k + 4) >> 1) + h];    // cols k+4+2h, k+5+2h
    v2f a0, a1;
    a0[0] = __expf(v0[0] - mx);  a0[1] = __expf(v0[1] - mx);
    a1[0] = __expf(v1[0] - mx);  a1[1] = __expf(v1[1] - mx);
    acc0 = __builtin_amdgcn_wmma_f32_16x16x4_f32(false, a0, false, bones,
                                                 (short)0, acc0, false, false);
    acc1 = __builtin_amdgcn_wmma_f32_16x16x4_f32(false, a1, false, bones,
                                                 (short)0, acc1, false, false);
  }
  v8f acc = acc0 + acc1;

  // D layout: VGPR j, lanes 0-15 -> row j, lanes 16-31 -> row j+8 (all N equal)
  __shared__ float wsum[8][16];
  if (m == 0) {  // lanes 0 and 16 carry rows 0-7 / 8-15 respectively
#pragma unroll
    for (int j = 0; j < 8; ++j) wsum[wave][h * 8 + j] = acc[j];
  }
  __syncthreads();
  if (lane < 16) {
    lse[R + lane] = mx + __logf(wsum[wave][lane]);
  }
}

// ---------------------------------------------------------------------------
// Kernel 2: CTC forward recursion. One block per batch element.
// Thread s owns lattice state s; alpha double-buffered in LDS (one barrier
// per step); emission gathers prefetched NB steps ahead with async-to-LDS.
// ---------------------------------------------------------------------------
__global__ __launch_bounds__(256) void ctc_forward_kernel(
    const float* __restrict__ x, const int* __restrict__ tgt,
    const float* __restrict__ lse, float* __restrict__ lossb) {
  const int b   = blockIdx.x;
  const int tid = threadIdx.x;

  __shared__ float alpha[2][224];
  __shared__ float ebuf[NB][224];
  __shared__ float lrow[T_];

  // stage this batch's lse row (8KB) into LDS once
  for (int i = tid; i < T_; i += 256) lrow[i] = lse[b * T_ + i];
  __syncthreads();

  int          lab   = 0;
  bool         skip  = false;
  const float* gbase = nullptr;

  if (tid < S_) {
    if (tid & 1) {
      const int q = tid >> 1;
      lab  = tgt[b * L_ + q];
      skip = (tid >= 3) && (lab != tgt[b * L_ + q - 1]);
    }
    gbase = x + ((size_t)b * T_) * C_ + lab;

    // alpha_0: only states 0 (blank) and 1 (first label) are reachable
    alpha[0][tid] = (tid < 2) ? (gbase[0] - lrow[0]) : NEG;

    // prime the async ring with steps t = 1 .. NB
#pragma unroll
    for (int t = 1; t <= NB; ++t)
      async_copy_b32(gbase + (size_t)t * C_, &ebuf[t & (NB - 1)][tid]);
  }
  __syncthreads();

  int p = 0;
  for (int t = 1; t < T_; ++t) {
    if (tid < S_) {
      // ensure the ring slot for step t has landed (tail: drain fully)
      if (t <= T_ - NB) wait_asynccnt_le<NB - 1>();
      else              wait_asynccnt_le<0>();

      const float e  = ebuf[t & (NB - 1)][tid] - lrow[t];
      const float a0 = alpha[p][tid];
      const float a1 = (tid >= 1) ? alpha[p][tid - 1] : NEG;
      const float a2 = skip ? alpha[p][tid - 2] : NEG;
      const float mm = fmaxf(a0, fmaxf(a1, a2));
      const float na = mm + __logf(__expf(a0 - mm) + __expf(a1 - mm) +
                                   __expf(a2 - mm)) + e;
      alpha[p ^ 1][tid] = na;
    }
    __syncthreads();
    if (tid < S_) {
      const int tp = t + NB;  // refill the slot just consumed
      if (tp < T_) async_copy_b32(gbase + (size_t)tp * C_,
                                  &ebuf[t & (NB - 1)][tid]);
    }
    p ^= 1;
  }

  if (tid == 0) {
    const float aN = alpha[p][S_ - 1];
    const float aM = alpha[p][S_ - 2];
    const float mm = fmaxf(aN, aM);
    const float ll = mm + __logf(__expf(aN - mm) + __expf(aM - mm));
    lossb[b] = -ll / (float)L_;
  }
}

// ---------------------------------------------------------------------------
// Kernel 3: mean over batch -> scalar loss
// ---------------------------------------------------------------------------
__global__ void ctc_reduce_kernel(const float* __restrict__ lossb,
                                  float* __restrict__ out) {
  float v = lossb[threadIdx.x];
#pragma unroll
  for (int off = 16; off > 0; off >>= 1) v += __shfl_down(v, off, 32);
  if (threadIdx.x == 0) out[0] = v / (float)B_;
}

// ---------------------------------------------------------------------------
extern "C" void kernel_launch(void* const* d_in, const int* in_sizes, int n_in,
                              void* d_out, int out_size, void* d_ws,
                              size_t ws_size, hipStream_t stream) {
  const float* x   = (const float*)d_in[0];   // [B,T,C] f32
  const int*   tgt = (const int*)d_in[1];     // [B,L] int
  float*       out = (float*)d_out;           // scalar f32

  float* lse   = (float*)d_ws;                // B*T floats
  float* lossb = lse + (size_t)B_ * T_;       // B floats

  lse_wmma_kernel<<<(B_ * T_) / 128, 256, 0, stream>>>(x, lse);
  ctc_forward_kernel<<<B_, 256, 0, stream>>>(x, tgt, lse, lossb);
  ctc_reduce_kernel<<<1, 32, 0, stream>>>(lossb, out);
}